// AdditiveAttention_66236985639003
// MI455X (gfx1250) — compile-verified
//
#include <hip/hip_runtime.h>
#include <hip/hip_bf16.h>

// ---------------------------------------------------------------------------
// AdditiveAttention on MI455X (gfx1250, wave32)
//   S=1024, T=64, B=8, KEY_DIM=QUERY_DIM=512, HIDDEN=256
//   phase 1: key_proj / query_proj GEMMs via V_WMMA_F32_16X16X4_F32
//   phase 2: fused tanh-dot + softmax, one 1024-thread block per (t,b)
// ---------------------------------------------------------------------------

#define S_LEN 1024
#define T_LEN 64
#define B_LEN 8
#define D_LEN 512
#define H_LEN 256

typedef __attribute__((ext_vector_type(2))) float v2f;
typedef __attribute__((ext_vector_type(8))) float v8f;

#if defined(__has_builtin)
#if __has_builtin(__builtin_amdgcn_tanhf)
#define TANH_F32(x) __builtin_amdgcn_tanhf(x)
#endif
#endif
#ifndef TANH_F32
#define TANH_F32(x) tanhf(x)
#endif

// ---------------------------------------------------------------------------
// Phase 1: Y[M,N] = X[M,D] @ W[D,N] + bias[N]   (f32, WMMA 16x16x4)
// One wave per 16x16 output tile; 8 waves (256 threads) per block.
//
// A-matrix VGPR layout (32-bit, 16x4 MxK): lanes 0-15 -> M=lane, {K, K+1};
//   lanes 16-31 -> M=lane-16, {K+2, K+3}. B mirrors with rows striped across
//   lanes within a VGPR. C/D: VGPR r = row r (lanes 0-15) / row r+8 (16-31).
// ---------------------------------------------------------------------------
__global__ __launch_bounds__(256) void
proj_gemm_wmma_f32(const float* __restrict__ X, const float* __restrict__ W,
                   const float* __restrict__ bias, float* __restrict__ Y,
                   int M, int D, int N) {
  const int wave = threadIdx.x >> 5;
  const int lane = threadIdx.x & 31;
  const int tilesN = N >> 4;
  const int tile = blockIdx.x * (blockDim.x >> 5) + wave;
  const int tM = tile / tilesN;
  const int tN = tile - tM * tilesN;
  if (tM * 16 >= M) return;  // wave-uniform: EXEC stays all-ones for WMMA

  const int half = lane >> 4;          // 0: lanes 0-15, 1: lanes 16-31
  const int l15  = lane & 15;
  const int row  = tM * 16 + l15;      // A row owned by this lane
  const int col  = tN * 16 + l15;      // B/C column owned by this lane
  const int khalf = half << 1;         // +0 or +2 within the K=4 step

  const float* __restrict__ xrow = X + (size_t)row * D;

  v8f c = {};
  for (int k = 0; k < D; k += 4) {
    const int kk = k + khalf;
    v2f a, b;
    a.x = xrow[kk];
    a.y = xrow[kk + 1];
    b.x = W[(size_t)kk * N + col];
    b.y = W[(size_t)(kk + 1) * N + col];
    c = __builtin_amdgcn_wmma_f32_16x16x4_f32(
        /*neg_a=*/false, a, /*neg_b=*/false, b,
        /*c_mod=*/(short)0, c, /*reuse_a=*/false, /*reuse_b=*/false);
  }

  const float bb = bias[col];
  const int m0 = tM * 16 + (half << 3);  // lanes 16-31 hold rows +8
#pragma unroll
  for (int r = 0; r < 8; ++r) {
    Y[(size_t)(m0 + r) * N + col] = c[r] + bb;
  }
}

// ---------------------------------------------------------------------------
// Phase 2: scores + softmax, fused.
//   block = (b, t), 1024 threads, thread s owns score[t,s,b] in a register.
//   qp[t,b,:] and v[:] packed as float2 in LDS (broadcast reads, no conflicts)
//   kp streamed as float4 (global_load_b128), hot in L2 (8MB total).
// ---------------------------------------------------------------------------
__global__ __launch_bounds__(1024) void
attn_tanh_softmax(const float* __restrict__ kp,   // (S,B,H) key_proj
                  const float* __restrict__ qp,   // (T,B,H) query_proj
                  const float* __restrict__ v,    // (H)
                  float* __restrict__ out) {      // (B,T,S)
  __shared__ float2 qv[H_LEN];
  __shared__ float red[40];

  const int b = blockIdx.x;
  const int t = blockIdx.y;
  const int s = threadIdx.x;

  if (threadIdx.x < H_LEN) {
    qv[threadIdx.x] = make_float2(qp[(size_t)(t * B_LEN + b) * H_LEN + threadIdx.x],
                                  v[threadIdx.x]);
  }
  __syncthreads();

  const float4* __restrict__ kp4 =
      (const float4*)(kp + (size_t)(s * B_LEN + b) * H_LEN);

  float acc = 0.f;
#pragma unroll 8
  for (int j = 0; j < H_LEN / 4; ++j) {
    const float4 k4 = kp4[j];
    float2 q;
    q = qv[4 * j + 0]; acc = fmaf(q.y, TANH_F32(k4.x + q.x), acc);
    q = qv[4 * j + 1]; acc = fmaf(q.y, TANH_F32(k4.y + q.x), acc);
    q = qv[4 * j + 2]; acc = fmaf(q.y, TANH_F32(k4.z + q.x), acc);
    q = qv[4 * j + 3]; acc = fmaf(q.y, TANH_F32(k4.w + q.x), acc);
  }

  const int wid = threadIdx.x >> 5;
  const int lid = threadIdx.x & 31;

  // ---- block max over 1024 scores (wave32 shuffle + LDS across 32 waves)
  float m = acc;
#pragma unroll
  for (int off = 16; off; off >>= 1) m = fmaxf(m, __shfl_xor(m, off, 32));
  if (lid == 0) red[wid] = m;
  __syncthreads();
  if (wid == 0) {
    float x = red[lid];
#pragma unroll
    for (int off = 16; off; off >>= 1) x = fmaxf(x, __shfl_xor(x, off, 32));
    if (lid == 0) red[36] = x;
  }
  __syncthreads();
  const float gmax = red[36];

  // ---- block sum of exp
  const float e = __expf(acc - gmax);
  float ssum = e;
#pragma unroll
  for (int off = 16; off; off >>= 1) ssum += __shfl_xor(ssum, off, 32);
  if (lid == 0) red[wid] = ssum;
  __syncthreads();
  if (wid == 0) {
    float x = red[lid];
#pragma unroll
    for (int off = 16; off; off >>= 1) x += __shfl_xor(x, off, 32);
    if (lid == 0) red[37] = x;
  }
  __syncthreads();

  const float inv = __frcp_rn(red[37]);
  out[((size_t)(b * T_LEN + t)) * S_LEN + s] = e * inv;
}

// ---------------------------------------------------------------------------
extern "C" void kernel_launch(void* const* d_in, const int* in_sizes, int n_in,
                              void* d_out, int out_size, void* d_ws, size_t ws_size,
                              hipStream_t stream) {
  (void)in_sizes; (void)n_in; (void)out_size; (void)ws_size;

  const float* key   = (const float*)d_in[0];  // (S,B,512)
  const float* query = (const float*)d_in[1];  // (T,B,512)
  const float* Wk    = (const float*)d_in[2];  // (512,256)
  const float* bk    = (const float*)d_in[3];  // (256)
  const float* Wq    = (const float*)d_in[4];  // (512,256)
  const float* bq    = (const float*)d_in[5];  // (256)
  const float* v     = (const float*)d_in[6];  // (256)
  float* out = (float*)d_out;                  // (B,T,S)

  float* kproj = (float*)d_ws;                                   // 8192*256 f32
  float* qproj = kproj + (size_t)S_LEN * B_LEN * H_LEN;          //  512*256 f32

  // key_proj: M = S*B = 8192 rows -> 8192/16 * 256/16 = 8192 tiles / 8 waves
  {
    const int Mk = S_LEN * B_LEN;
    const int tiles = (Mk / 16) * (H_LEN / 16);
    proj_gemm_wmma_f32<<<tiles / 8, 256, 0, stream>>>(key, Wk, bk, kproj,
                                                      Mk, D_LEN, H_LEN);
  }
  // query_proj: M = T*B = 512 rows -> 512 tiles / 8 waves
  {
    const int Mq = T_LEN * B_LEN;
    const int tiles = (Mq / 16) * (H_LEN / 16);
    proj_gemm_wmma_f32<<<tiles / 8, 256, 0, stream>>>(query, Wq, bq, qproj,
                                                      Mq, D_LEN, H_LEN);
  }
  // fused tanh-dot + softmax: one block per (b, t)
  attn_tanh_softmax<<<dim3(B_LEN, T_LEN), 1024, 0, stream>>>(kproj, qproj, v, out);
}